// SparseLocalCrossAttention_72791105733143
// MI455X (gfx1250) — compile-verified
//
#include <hip/hip_runtime.h>
#include <hip/hip_bf16.h>
#include <math.h>

// ---------------------------------------------------------------------------
// SparseLocalCrossAttention for MI455X (gfx1250), f32 WMMA 16x16x4 path.
// D=768, T=256 (text tokens), I=784 (image tokens). All dims multiples of 16.
// All GEMM shapes static -> tile count & B strides resolved at compile time
// so the k-loop is branch-free straight-line WMMA code.
// ---------------------------------------------------------------------------

typedef float v2f __attribute__((ext_vector_type(2)));
typedef float v8f __attribute__((ext_vector_type(8)));

#define D_DIM 768
#define T_TOK 256
#define I_TOK 784
#define LN_EPS 1e-5f

__device__ __forceinline__ float sigmoidf_(float x) {
    return 1.0f / (1.0f + __expf(-x));
}

// ---------------------------------------------------------------------------
// Core WMMA strip: one wave computes a 16(M) x (NT*16)(N) strip of C = A*B.
// A row-major (k contiguous, row stride lda).
// BT=false: B row-major [k][n], row stride D_DIM (compile-time).
// BT=true : B is K^T, i.e. element (k,n) = B[n*D_DIM + k] (k contiguous ->
//           b-pair loads become single b64 loads).
// Fragment layout per CDNA5 ISA (V_WMMA_F32_16X16X4_F32):
//   A: lane(l16,half): a.x=A[m][k0+2*half], a.y=A[m][k0+2*half+1]
//   B: lane(l16,half): b.x=B[k0+2*half][n], b.y=B[k0+2*half+1][n]
//   C/D: vgpr r, lane(l16,half): C[m0+half*8+r][n0+l16]
// k unrolled x2: all loads of an 8-k chunk are issued before its WMMAs.
// ---------------------------------------------------------------------------
template<int NT, bool BT>
__device__ __forceinline__ void wmma_strip(
    const float* __restrict__ A, int lda,
    const float* __restrict__ B,
    int m0, int n0, int K,
    v8f acc[NT])
{
    constexpr size_t SBK = BT ? 1 : D_DIM;   // stride along k
    constexpr size_t SBN = BT ? D_DIM : 1;   // stride along n

    const int lane = threadIdx.x & 31;
    const int half = lane >> 4;      // 0: k0,k0+1   1: k0+2,k0+3
    const int l16  = lane & 15;

    const float* Ap = A + (size_t)(m0 + l16) * lda + 2 * half;
    const float* Bp = B + (size_t)(n0 + l16) * SBN + (size_t)(2 * half) * SBK;

    for (int k0 = 0; k0 < K; k0 += 8) {
        v2f a0 = *(const v2f*)(Ap + k0);
        v2f a1 = *(const v2f*)(Ap + k0 + 4);
        v2f b0[NT], b1[NT];
#pragma unroll
        for (int t = 0; t < NT; ++t) {
            const float* Bt = Bp + (size_t)(t * 16) * SBN;
            if (BT) {
                b0[t] = *(const v2f*)(Bt + k0);       // contiguous along k
                b1[t] = *(const v2f*)(Bt + k0 + 4);
            } else {
                b0[t].x = Bt[(size_t)(k0    ) * SBK];
                b0[t].y = Bt[(size_t)(k0 + 1) * SBK];
                b1[t].x = Bt[(size_t)(k0 + 4) * SBK];
                b1[t].y = Bt[(size_t)(k0 + 5) * SBK];
            }
        }
#pragma unroll
        for (int t = 0; t < NT; ++t)
            acc[t] = __builtin_amdgcn_wmma_f32_16x16x4_f32(
                false, a0, false, b0[t], (short)0, acc[t], false, false);
#pragma unroll
        for (int t = 0; t < NT; ++t)
            acc[t] = __builtin_amdgcn_wmma_f32_16x16x4_f32(
                false, a1, false, b1[t], (short)0, acc[t], false, false);
    }
}

// ---------------------------------------------------------------------------
// Generic GEMM: C[M,N] = A[M,K] * B(row-major, ld=768) + bias (per column,
// may be null). Block = 256 threads = 8 waves stacked along M; each block
// covers NT*16 columns starting at n_base + blockIdx.x * NT*16.
// ---------------------------------------------------------------------------
template<int NT>
__global__ void gemm_wmma_kernel(
    const float* __restrict__ A, int lda,
    const float* __restrict__ B,
    const float* __restrict__ bias,
    float* __restrict__ C, int ldc,
    int M, int n_base, int K)
{
    const int wave = threadIdx.x >> 5;
    const int m0 = (blockIdx.y * 8 + wave) * 16;
    const int n0 = n_base + blockIdx.x * (NT * 16);
    if (m0 >= M) return;

    const v8f z = {0.f,0.f,0.f,0.f,0.f,0.f,0.f,0.f};
    v8f acc[NT];
#pragma unroll
    for (int t = 0; t < NT; ++t) acc[t] = z;

    wmma_strip<NT, false>(A, lda, B, m0, n0, K, acc);

    const int lane = threadIdx.x & 31;
    const int half = lane >> 4;
    const int l16  = lane & 15;
#pragma unroll
    for (int t = 0; t < NT; ++t) {
        const int col = n0 + t * 16 + l16;
        const float bv = bias ? bias[col] : 0.0f;
#pragma unroll
        for (int r = 0; r < 8; ++r) {
            C[(size_t)(m0 + half * 8 + r) * ldc + col] = acc[t][r] + bv;
        }
    }
}

// ---------------------------------------------------------------------------
// Scores GEMM with fused epilogue:
//   probs[i,j] = sigmoid( (Q[i]·K[j]) * mask[i,j] / sqrt(D) )
// B = K^T (BT=true): per-lane b-pairs are contiguous k -> b64 loads.
// N=784 handled as 12 blocks of NT=4 plus one NT=1 launch at n_base=768.
// ---------------------------------------------------------------------------
template<int NT>
__global__ void scores_probs_kernel(
    const float* __restrict__ Q,
    const float* __restrict__ Km,
    const float* __restrict__ mask,
    float* __restrict__ probs,
    int n_base)
{
    const int wave = threadIdx.x >> 5;
    const int m0 = (blockIdx.y * 8 + wave) * 16;
    const int n0 = n_base + blockIdx.x * (NT * 16);
    if (m0 >= T_TOK) return;

    const v8f z = {0.f,0.f,0.f,0.f,0.f,0.f,0.f,0.f};
    v8f acc[NT];
#pragma unroll
    for (int t = 0; t < NT; ++t) acc[t] = z;

    wmma_strip<NT, true>(Q, D_DIM, Km, m0, n0, D_DIM, acc);

    const int lane = threadIdx.x & 31;
    const int half = lane >> 4;
    const int l16  = lane & 15;
    const float scale = 0.03608439182435161f;   // 1/sqrt(768)
#pragma unroll
    for (int t = 0; t < NT; ++t) {
        const int col = n0 + t * 16 + l16;
#pragma unroll
        for (int r = 0; r < 8; ++r) {
            const int row = m0 + half * 8 + r;
            const float s = acc[t][r] * mask[(size_t)row * I_TOK + col] * scale;
            probs[(size_t)row * I_TOK + col] = sigmoidf_(s);
        }
    }
}

// ---------------------------------------------------------------------------
// Fused pairwise mask kernel (never materializes the 616MB h tensor):
//   mask[i,j] = sigmoid( sum_d relu(HQ[i,d] + HK[j,d]) * Wm2[d] + bm2 )
// Block handles a 16x16 (i,j) tile; HQ/HK streamed through LDS in 64-wide
// chunks. Row pitch 68 floats: keeps float4 stores 16B-aligned and makes
// bank(r,d) = (4r+d)%64 distinct across rows (conflict-free).
// ---------------------------------------------------------------------------
__global__ void mask_kernel(
    const float* __restrict__ HQ,
    const float* __restrict__ HK,
    const float* __restrict__ Wm2,
    const float* __restrict__ bm2,
    float* __restrict__ mask)
{
    __shared__ float sHQ[16][68];
    __shared__ float sHK[16][68];
    __shared__ float sW[64];

    const int tid  = threadIdx.x;
    const int i    = tid & 15;          // text-token within tile
    const int j    = tid >> 4;          // image-token within tile
    const int i0   = blockIdx.y * 16;
    const int j0   = blockIdx.x * 16;
    const int lrow = tid >> 4;          // cooperative-load row 0..15
    const int lcol = (tid & 15) * 4;    // cooperative-load col (float4)

    float ml = 0.0f;
    for (int c = 0; c < D_DIM; c += 64) {
        __syncthreads();   // protect LDS from previous iteration's readers
        *(float4*)&sHQ[lrow][lcol] =
            *(const float4*)&HQ[(size_t)(i0 + lrow) * D_DIM + c + lcol];
        *(float4*)&sHK[lrow][lcol] =
            *(const float4*)&HK[(size_t)(j0 + lrow) * D_DIM + c + lcol];
        if (tid < 16)
            *(float4*)&sW[tid * 4] = *(const float4*)&Wm2[c + tid * 4];
        __syncthreads();
#pragma unroll 8
        for (int d = 0; d < 64; ++d) {
            ml += fmaxf(sHQ[i][d] + sHK[j][d], 0.0f) * sW[d];
        }
    }
    mask[(size_t)(i0 + i) * I_TOK + (j0 + j)] = sigmoidf_(ml + bm2[0]);
}

// ---------------------------------------------------------------------------
// LayerNorm: one block (256 threads) per row of 768.
// ---------------------------------------------------------------------------
__global__ void layernorm_kernel(
    const float* __restrict__ X,
    const float* __restrict__ g,
    const float* __restrict__ b,
    float* __restrict__ Y)
{
    __shared__ float s1[256];
    __shared__ float s2[256];
    const int row = blockIdx.x;
    const int tid = threadIdx.x;
    const float* x = X + (size_t)row * D_DIM;

    const float v0 = x[tid];
    const float v1 = x[tid + 256];
    const float v2 = x[tid + 512];
    s1[tid] = v0 + v1 + v2;
    s2[tid] = v0 * v0 + v1 * v1 + v2 * v2;
    __syncthreads();
    for (int s = 128; s > 0; s >>= 1) {
        if (tid < s) { s1[tid] += s1[tid + s]; s2[tid] += s2[tid + s]; }
        __syncthreads();
    }
    const float mu  = s1[0] * (1.0f / 768.0f);
    const float var = s2[0] * (1.0f / 768.0f) - mu * mu;
    const float inv = rsqrtf(var + LN_EPS);

    float* y = Y + (size_t)row * D_DIM;
    y[tid]       = (v0 - mu) * inv * g[tid]       + b[tid];
    y[tid + 256] = (v1 - mu) * inv * g[tid + 256] + b[tid + 256];
    y[tid + 512] = (v2 - mu) * inv * g[tid + 512] + b[tid + 512];
}

// ---------------------------------------------------------------------------
// Deterministic single-block reduction for sparsity loss = mean(|mask|).
// ---------------------------------------------------------------------------
__global__ void loss_kernel(const float* __restrict__ mask,
                            float* __restrict__ out)
{
    __shared__ float s1[256];
    const int tid = threadIdx.x;
    float acc = 0.0f;
    for (int idx = tid; idx < T_TOK * I_TOK; idx += 256)
        acc += fabsf(mask[idx]);
    s1[tid] = acc;
    __syncthreads();
    for (int s = 128; s > 0; s >>= 1) {
        if (tid < s) s1[tid] += s1[tid + s];
        __syncthreads();
    }
    if (tid == 0) out[0] = s1[0] * (1.0f / (float)(T_TOK * I_TOK));
}

// ---------------------------------------------------------------------------
extern "C" void kernel_launch(void* const* d_in, const int* in_sizes, int n_in,
                              void* d_out, int out_size, void* d_ws, size_t ws_size,
                              hipStream_t stream) {
    const float* image = (const float*)d_in[0];
    const float* text  = (const float*)d_in[1];
    const float* Wq  = (const float*)d_in[2];
    const float* bq  = (const float*)d_in[3];
    const float* Wk  = (const float*)d_in[4];
    const float* bk  = (const float*)d_in[5];
    const float* Wv  = (const float*)d_in[6];
    const float* bv  = (const float*)d_in[7];
    const float* Wm1 = (const float*)d_in[8];   // (1536,768): rows [0,768)=Wm1_q, [768,1536)=Wm1_k
    const float* bm1 = (const float*)d_in[9];
    const float* Wm2 = (const float*)d_in[10];  // (768,1)
    const float* bm2 = (const float*)d_in[11];  // (1,)
    const float* Wo  = (const float*)d_in[12];
    const float* bo  = (const float*)d_in[13];
    const float* ln_g = (const float*)d_in[14];
    const float* ln_b = (const float*)d_in[15];

    float* out      = (float*)d_out;
    float* outCtx   = out;                                  // 256*768
    float* outProbs = out + T_TOK * D_DIM;                  // 256*784
    float* outLoss  = out + T_TOK * D_DIM + T_TOK * I_TOK;  // 1

    // Workspace layout (floats); total 2,400,256 floats ~= 9.6 MB.
    float* ws  = (float*)d_ws;
    float* Q    = ws;                          // 256*768 = 196608
    float* Km   = Q   + T_TOK * D_DIM;         // 784*768 = 602112
    float* V    = Km  + I_TOK * D_DIM;         // 602112
    float* HQ   = V   + I_TOK * D_DIM;         // 196608
    float* HK   = HQ  + T_TOK * D_DIM;         // 602112
    float* MASK = HK  + I_TOK * D_DIM;         // 256*784 = 200704
    float* CTX1 = HQ;   // HQ dead after mask_kernel; reused for probs@V
    float* CTX2 = Q;    // Q dead after scores_probs_kernel; reused for @Wo

    const dim3 blk(256);   // 8 wave32s

    // q = text @ Wq + bq                       (256x768, K=768)
    gemm_wmma_kernel<4><<<dim3(12, 2), blk, 0, stream>>>(
        text, D_DIM, Wq, bq, Q, D_DIM, T_TOK, 0, D_DIM);
    // k = image @ Wk + bk                      (784x768)
    gemm_wmma_kernel<4><<<dim3(12, 7), blk, 0, stream>>>(
        image, D_DIM, Wk, bk, Km, D_DIM, I_TOK, 0, D_DIM);
    // v = image @ Wv + bv                      (784x768)
    gemm_wmma_kernel<4><<<dim3(12, 7), blk, 0, stream>>>(
        image, D_DIM, Wv, bv, V, D_DIM, I_TOK, 0, D_DIM);
    // hq = q @ Wm1_q + bm1                     (256x768)
    gemm_wmma_kernel<4><<<dim3(12, 2), blk, 0, stream>>>(
        Q, D_DIM, Wm1, bm1, HQ, D_DIM, T_TOK, 0, D_DIM);
    // hk = k @ Wm1_k                           (784x768, no bias)
    gemm_wmma_kernel<4><<<dim3(12, 7), blk, 0, stream>>>(
        Km, D_DIM, Wm1 + (size_t)D_DIM * D_DIM, nullptr, HK, D_DIM, I_TOK, 0, D_DIM);
    // mask[i,j] = sigmoid(sum_d relu(hq+hk)*Wm2 + bm2)   (fused, no h tensor)
    mask_kernel<<<dim3(49, 16), blk, 0, stream>>>(HQ, HK, Wm2, bm2, MASK);
    // probs = sigmoid(q @ k.T * mask / sqrt(D))  -> directly into d_out
    scores_probs_kernel<4><<<dim3(12, 2), blk, 0, stream>>>(Q, Km, MASK, outProbs, 0);
    scores_probs_kernel<1><<<dim3(1, 2), blk, 0, stream>>>(Q, Km, MASK, outProbs, 768);
    // ctx1 = probs @ v                         (256x768, K=784)
    gemm_wmma_kernel<4><<<dim3(12, 2), blk, 0, stream>>>(
        outProbs, I_TOK, V, nullptr, CTX1, D_DIM, T_TOK, 0, I_TOK);
    // ctx2 = ctx1 @ Wo + bo                    (256x768)
    gemm_wmma_kernel<4><<<dim3(12, 2), blk, 0, stream>>>(
        CTX1, D_DIM, Wo, bo, CTX2, D_DIM, T_TOK, 0, D_DIM);
    // layernorm -> d_out ctx
    layernorm_kernel<<<dim3(T_TOK), blk, 0, stream>>>(CTX2, ln_g, ln_b, outCtx);
    // sparsity loss -> d_out tail (deterministic single-block reduce)
    loss_kernel<<<dim3(1), blk, 0, stream>>>(MASK, outLoss);
}